// HyperMLP_38448547234254
// MI455X (gfx1250) — compile-verified
//
#include <hip/hip_runtime.h>

// Problem constants (match reference)
#define BDIM   16
#define SDIM   256
#define INDIM  128
#define OUTDIM 128
#define HIDDIM 512
#define BS     (BDIM * SDIM)      // 4096 tokens
#define KBIG   (HIDDIM * INDIM)   // 65536 (reduction dim of GEMM3)
#define KCHUNKS 256
#define KC     (KBIG / KCHUNKS)   // 256 per split-K chunk

typedef __attribute__((ext_vector_type(2))) float v2f;
typedef __attribute__((ext_vector_type(8))) float v8f;

// fp32 WMMA: D(16x16) = A(16x4) * B(4x16) + C, full fp32 (matches reference dtype)
__device__ __forceinline__ v8f wmma4(v2f a, v2f b, v8f c) {
    return __builtin_amdgcn_wmma_f32_16x16x4_f32(
        /*neg_a=*/false, a, /*neg_b=*/false, b,
        /*c_mod=*/(short)0, c, /*reuse_a=*/false, /*reuse_b=*/false);
}

// --------------------------------------------------------------------------
// GEMM1: H[4096,512] = relu( X[4096,128] @ W1[128,512] + b1 )
// One wave computes a 16x64 output slab (4 N-tiles share the A fragment).
// --------------------------------------------------------------------------
__global__ __launch_bounds__(32)
void k_gemm1(const float* __restrict__ X, const float* __restrict__ W1,
             const float* __restrict__ b1, float* __restrict__ H) {
    const int lane = threadIdx.x;
    const int lm = lane & 15;        // row/col within tile
    const int lh = lane >> 4;        // K-pair selector (0 or 1)
    const int rowA = blockIdx.x * 16 + lm;
    const int colBase = blockIdx.y * 64;

    v8f acc[4] = {};
    for (int k = 0; k < INDIM; k += 4) {
        const int kb = k + 2 * lh;
        v2f a = *(const v2f*)(X + (size_t)rowA * INDIM + kb);   // contiguous b64
        #pragma unroll
        for (int j = 0; j < 4; ++j) {
            const int col = colBase + j * 16 + lm;
            v2f b;
            b.x = W1[(size_t)kb * HIDDIM + col];
            b.y = W1[(size_t)(kb + 1) * HIDDIM + col];
            acc[j] = wmma4(a, b, acc[j]);
        }
    }
    #pragma unroll
    for (int j = 0; j < 4; ++j) {
        const int col = colBase + j * 16 + lm;
        const float bias = b1[col];
        #pragma unroll
        for (int r = 0; r < 8; ++r) {
            const int orow = blockIdx.x * 16 + r + 8 * lh;
            const float v = acc[j][r] + bias;
            H[(size_t)orow * HIDDIM + col] = v > 0.f ? v : 0.f;   // relu
        }
    }
}

// --------------------------------------------------------------------------
// GEMM2 (batched): M[b][512,128] = H[b]^T (512x256) @ X[b] (256x128)
// One wave computes a full 16x128 row slab (8 N-tiles reuse the A fragment).
// --------------------------------------------------------------------------
__global__ __launch_bounds__(32)
void k_gemm2(const float* __restrict__ X, const float* __restrict__ H,
             float* __restrict__ M) {
    const int lane = threadIdx.x;
    const int lm = lane & 15;
    const int lh = lane >> 4;
    const int b = blockIdx.x;
    const int hRow = blockIdx.y * 16 + lm;

    const float* Xb = X + (size_t)b * SDIM * INDIM;
    const float* Hb = H + (size_t)b * SDIM * HIDDIM;
    float*       Mb = M + (size_t)b * HIDDIM * INDIM;

    v8f acc[8] = {};
    for (int k = 0; k < SDIM; k += 4) {       // K = S = 256
        const int kb = k + 2 * lh;
        v2f a;                                 // A[m=h][k=s] = H[s][h] (transposed read)
        a.x = Hb[(size_t)kb * HIDDIM + hRow];
        a.y = Hb[(size_t)(kb + 1) * HIDDIM + hRow];
        #pragma unroll
        for (int j = 0; j < 8; ++j) {
            const int col = j * 16 + lm;
            v2f bf;
            bf.x = Xb[(size_t)kb * INDIM + col];
            bf.y = Xb[(size_t)(kb + 1) * INDIM + col];
            acc[j] = wmma4(a, bf, acc[j]);
        }
    }
    #pragma unroll
    for (int j = 0; j < 8; ++j) {
        const int col = j * 16 + lm;
        #pragma unroll
        for (int r = 0; r < 8; ++r) {
            const int row = blockIdx.y * 16 + r + 8 * lh;
            Mb[(size_t)row * INDIM + col] = acc[j][r];
        }
    }
}

// --------------------------------------------------------------------------
// GEMM3 (split-K): P[chunk][16,128] = Mflat(16x65536)[:,chunk] @ W2(65536x128)[chunk,:]
// Dominant kernel: streams W2 (33.5 MB) exactly once.
// --------------------------------------------------------------------------
__global__ __launch_bounds__(32)
void k_gemm3(const float* __restrict__ Mflat, const float* __restrict__ W2,
             float* __restrict__ P) {
    const int lane = threadIdx.x;
    const int lm = lane & 15;
    const int lh = lane >> 4;
    const int col = blockIdx.y * 16 + lm;
    const int k0 = blockIdx.x * KC;

    v8f acc = {};
    for (int kk = 0; kk < KC; kk += 4) {
        const int kb = k0 + kk + 2 * lh;
        v2f a = *(const v2f*)(Mflat + (size_t)lm * KBIG + kb);   // rows = 16 batches
        v2f bf;
        bf.x = W2[(size_t)kb * OUTDIM + col];
        bf.y = W2[(size_t)(kb + 1) * OUTDIM + col];
        acc = wmma4(a, bf, acc);
    }
    float* Pc = P + (size_t)blockIdx.x * (BDIM * OUTDIM);
    #pragma unroll
    for (int r = 0; r < 8; ++r) {
        const int row = r + 8 * lh;
        Pc[(size_t)row * OUTDIM + col] = acc[r];
    }
}

// Deterministic split-K reduction: Y0[e] = sum_c P[c][e]
__global__ void k_reduce(const float* __restrict__ P, float* __restrict__ Y0) {
    const int e = blockIdx.x * blockDim.x + threadIdx.x;   // 0..2047
    float s = 0.f;
    for (int c = 0; c < KCHUNKS; ++c) s += P[(size_t)c * (BDIM * OUTDIM) + e];
    Y0[e] = s;
}

// xsum[b,i] = sum_s x[b,s,i]   (for the b2 bias term)
__global__ void k_xsum(const float* __restrict__ X, float* __restrict__ xsum) {
    const int b = blockIdx.x, i = threadIdx.x;
    const float* Xb = X + (size_t)b * SDIM * INDIM;
    float s = 0.f;
    for (int t = 0; t < SDIM; ++t) s += Xb[(size_t)t * INDIM + i];
    xsum[b * INDIM + i] = s;
}

// T[b,o'] = Y0[b,o'] + sum_i xsum[b,i]*b2[i*128+o'];  out[b,o] = T[b,:] @ Wfc[:,o] + bfc[o]
__global__ void k_final(const float* __restrict__ Y0, const float* __restrict__ xsum,
                        const float* __restrict__ b2, const float* __restrict__ Wfc,
                        const float* __restrict__ bfc, float* __restrict__ out) {
    __shared__ float t[OUTDIM];
    const int b = blockIdx.x, o = threadIdx.x;
    float acc = Y0[b * OUTDIM + o];
    for (int i = 0; i < INDIM; ++i)
        acc += xsum[b * INDIM + i] * b2[(size_t)i * OUTDIM + o];
    t[o] = acc;
    __syncthreads();
    float r = bfc[o];
    for (int op = 0; op < OUTDIM; ++op)
        r += t[op] * Wfc[(size_t)op * OUTDIM + o];
    out[b * OUTDIM + o] = r;
}

extern "C" void kernel_launch(void* const* d_in, const int* in_sizes, int n_in,
                              void* d_out, int out_size, void* d_ws, size_t ws_size,
                              hipStream_t stream) {
    const float* x   = (const float*)d_in[0];   // [16,256,128]
    const float* W1  = (const float*)d_in[1];   // [128,512]
    const float* b1  = (const float*)d_in[2];   // [512]
    const float* W2  = (const float*)d_in[3];   // [512,16384] == (65536,128) over (h*128+i, o)
    const float* b2  = (const float*)d_in[4];   // [16384]
    const float* Wfc = (const float*)d_in[5];   // [128,128]
    const float* bfc = (const float*)d_in[6];   // [128]
    float* out = (float*)d_out;                 // [16,128]

    // Workspace layout (floats): total ~14.7 MB
    float* H    = (float*)d_ws;                              // 4096*512
    float* Mb   = H    + (size_t)BS * HIDDIM;                // 16*512*128
    float* P    = Mb   + (size_t)BDIM * HIDDIM * INDIM;      // 256*2048
    float* xs   = P    + (size_t)KCHUNKS * BDIM * OUTDIM;    // 2048
    float* Y0   = xs   + (size_t)BDIM * INDIM;               // 2048

    k_gemm1 <<<dim3(BS / 16, HIDDIM / 64), 32, 0, stream>>>(x, W1, b1, H);
    k_gemm2 <<<dim3(BDIM, HIDDIM / 16),    32, 0, stream>>>(x, H, Mb);
    k_gemm3 <<<dim3(KCHUNKS, OUTDIM / 16), 32, 0, stream>>>(Mb, W2, P);
    k_xsum  <<<BDIM, INDIM,                 0, stream>>>(x, xs);
    k_reduce<<<(BDIM * OUTDIM) / 256, 256,  0, stream>>>(P, Y0);
    k_final <<<BDIM, OUTDIM,                0, stream>>>(Y0, xs, b2, Wfc, bfc, out);
}